// TMLoss_15522011807776
// MI455X (gfx1250) — compile-verified
//
#include <hip/hip_runtime.h>
#include <math.h>

#define B_BATCH 2048
#define N_PTS   1024
#define EPS_TM  1e-8f

typedef __attribute__((ext_vector_type(2))) float v2f;
typedef __attribute__((ext_vector_type(8))) float v8f;

// ---------------------------------------------------------------------------
// Pass 1: per-batch augmented cross-covariance via V_WMMA_F32_16X16X4_F32.
// A (16x4) rows = components of actual (q), row 3 = ones.
// B (4x16)  cols = components of pred   (p), col 3 = ones.
// D[m][n] accumulates: [3x3] = sum q_m p_n, row3 = sum p, col3 = sum q.
// One workgroup (8 waves) per batch; 256-point chunks staged in LDS.
// ---------------------------------------------------------------------------
__global__ __launch_bounds__(256) void tm_pass1_cov(const float* __restrict__ pred,
                                                    const float* __restrict__ act,
                                                    float* __restrict__ ws0) {
  __shared__ float lds_p[256 * 3];
  __shared__ float lds_q[256 * 3];
  __shared__ float red[8 * 16];

  const int tid  = threadIdx.x;
  const int wave = tid >> 5;
  const int lane = tid & 31;
  const int c    = lane & 15;   // matrix row (A) / col (B) handled by this lane
  const int hi   = lane >> 4;
  const int b    = blockIdx.x;
  const unsigned base = (unsigned)b * (N_PTS * 3);

  const int  cc   = (c < 3) ? c : 0;            // clamped LDS component index
  const float fill = (c == 3) ? 1.0f : 0.0f;    // ones-row/col augmentation
  const bool cv   = (c < 3);

  v8f acc = {0.f, 0.f, 0.f, 0.f, 0.f, 0.f, 0.f, 0.f};

  for (int chunk = 0; chunk < N_PTS / 256; ++chunk) {
    __syncthreads();   // protect previous chunk's LDS reads
    const unsigned cbase = base + (unsigned)chunk * (256 * 3);
    for (int idx = tid; idx < 256 * 3; idx += 256) {
      lds_p[idx] = pred[cbase + idx];
      lds_q[idx] = act[cbase + idx];
    }
    __syncthreads();

    // wave handles 32 points: 8 WMMA steps of K=4
#pragma unroll
    for (int kk = 0; kk < 8; ++kk) {
      const int pbase = wave * 32 + kk * 4;
      // A/B 32-bit K=4 layout: lanes<16 {v0:K0, v1:K1}, lanes>=16 {v0:K2, v1:K3}
      const int pt0 = pbase + 2 * hi;
      float a0 = lds_q[pt0 * 3 + cc];
      float a1 = lds_q[(pt0 + 1) * 3 + cc];
      float b0 = lds_p[pt0 * 3 + cc];
      float b1 = lds_p[(pt0 + 1) * 3 + cc];
      a0 = cv ? a0 : fill;  a1 = cv ? a1 : fill;
      b0 = cv ? b0 : fill;  b1 = cv ? b1 : fill;
      v2f av = {a0, a1};
      v2f bv = {b0, b1};
      acc = __builtin_amdgcn_wmma_f32_16x16x4_f32(false, av, false, bv,
                                                  (short)0, acc, false, false);
    }
  }

  // D layout: VGPR m holds row m (m=0..7) for lanes 0-15 (lane = col n)
  if (lane < 4) {
    red[wave * 16 + 0 * 4 + lane] = acc[0];
    red[wave * 16 + 1 * 4 + lane] = acc[1];
    red[wave * 16 + 2 * 4 + lane] = acc[2];
    red[wave * 16 + 3 * 4 + lane] = acc[3];
  }
  __syncthreads();
  if (tid < 16) {
    float s = 0.f;
#pragma unroll
    for (int w = 0; w < 8; ++w) s += red[w * 16 + tid];
    ws0[(unsigned)b * 16 + tid] = s;   // 4x4 augmented covariance (row-major)
  }
}

// ---------------------------------------------------------------------------
// Pass 2: one thread per batch. Build centered M, eigensolve M^T M (Jacobi),
// R = M V diag(1/s0, 1/s1, d/s2) V^T  with d = sign(det M).
// Overwrites ws0[b*16 .. ] with {R[9], mu_p[3], mu_q[3]}.
// ---------------------------------------------------------------------------
__device__ __forceinline__ void jrot(float A[3][3], float V[3][3], int p, int q) {
  float apq = A[p][q];
  if (fabsf(apq) > 1e-30f) {
    float tau = (A[q][q] - A[p][p]) / (2.0f * apq);
    float t   = ((tau >= 0.f) ? 1.0f : -1.0f) / (fabsf(tau) + sqrtf(1.0f + tau * tau));
    float cth = 1.0f / sqrtf(1.0f + t * t);
    float sth = t * cth;
#pragma unroll
    for (int k = 0; k < 3; ++k) {
      float akp = A[k][p], akq = A[k][q];
      A[k][p] = cth * akp - sth * akq;
      A[k][q] = sth * akp + cth * akq;
    }
#pragma unroll
    for (int k = 0; k < 3; ++k) {
      float apk = A[p][k], aqk = A[q][k];
      A[p][k] = cth * apk - sth * aqk;
      A[q][k] = sth * apk + cth * aqk;
    }
#pragma unroll
    for (int k = 0; k < 3; ++k) {
      float vkp = V[k][p], vkq = V[k][q];
      V[k][p] = cth * vkp - sth * vkq;
      V[k][q] = sth * vkp + cth * vkq;
    }
  }
}

__global__ __launch_bounds__(256) void tm_pass2_rot(float* __restrict__ ws0) {
  int b = blockIdx.x * 256 + threadIdx.x;
  if (b >= B_BATCH) return;

  float v[16];
#pragma unroll
  for (int i = 0; i < 16; ++i) v[i] = ws0[(unsigned)b * 16 + i];

  const float invn = 1.0f / (float)N_PTS;
  float sumQ[3] = {v[3], v[7], v[11]};   // val[m][3]
  float sumP[3] = {v[12], v[13], v[14]}; // val[3][n]

  float M[3][3];
#pragma unroll
  for (int i = 0; i < 3; ++i)
#pragma unroll
    for (int j = 0; j < 3; ++j)
      M[i][j] = v[i * 4 + j] - sumQ[i] * sumP[j] * invn;

  // A = M^T M (symmetric)
  float A[3][3];
#pragma unroll
  for (int i = 0; i < 3; ++i)
#pragma unroll
    for (int j = 0; j < 3; ++j)
      A[i][j] = M[0][i] * M[0][j] + M[1][i] * M[1][j] + M[2][i] * M[2][j];

  float V[3][3] = {{1.f, 0.f, 0.f}, {0.f, 1.f, 0.f}, {0.f, 0.f, 1.f}};
#pragma unroll
  for (int sweep = 0; sweep < 6; ++sweep) {
    jrot(A, V, 0, 1);
    jrot(A, V, 0, 2);
    jrot(A, V, 1, 2);
  }
  float lam[3] = {A[0][0], A[1][1], A[2][2]};

  // sort eigenpairs descending (swap V columns alongside)
#define TM_SWAP_EIG(aa, bb)                                        \
  {                                                                \
    float tl = lam[aa]; lam[aa] = lam[bb]; lam[bb] = tl;           \
    for (int k = 0; k < 3; ++k) {                                  \
      float tv = V[k][aa]; V[k][aa] = V[k][bb]; V[k][bb] = tv;     \
    }                                                              \
  }
  if (lam[0] < lam[1]) TM_SWAP_EIG(0, 1);
  if (lam[0] < lam[2]) TM_SWAP_EIG(0, 2);
  if (lam[1] < lam[2]) TM_SWAP_EIG(1, 2);
#undef TM_SWAP_EIG

  float det = M[0][0] * (M[1][1] * M[2][2] - M[1][2] * M[2][1])
            - M[0][1] * (M[1][0] * M[2][2] - M[1][2] * M[2][0])
            + M[0][2] * (M[1][0] * M[2][1] - M[1][1] * M[2][0]);
  float d = (det >= 0.f) ? 1.0f : -1.0f;

  float invs[3];
#pragma unroll
  for (int j = 0; j < 3; ++j) {
    float s = sqrtf(fmaxf(lam[j], 0.f));
    invs[j] = (s > 1e-12f) ? (1.0f / s) : 0.f;
  }
  invs[2] *= d;  // det correction on smallest singular direction

  // W = (M V) * diag(invs)  == U * diag(1,1,d)
  float W[3][3];
#pragma unroll
  for (int i = 0; i < 3; ++i)
#pragma unroll
    for (int j = 0; j < 3; ++j)
      W[i][j] = (M[i][0] * V[0][j] + M[i][1] * V[1][j] + M[i][2] * V[2][j]) * invs[j];

  // R = W V^T
  float R[3][3];
#pragma unroll
  for (int i = 0; i < 3; ++i)
#pragma unroll
    for (int j = 0; j < 3; ++j)
      R[i][j] = W[i][0] * V[j][0] + W[i][1] * V[j][1] + W[i][2] * V[j][2];

  float* o = ws0 + (unsigned)b * 16;
#pragma unroll
  for (int i = 0; i < 3; ++i)
#pragma unroll
    for (int j = 0; j < 3; ++j)
      o[i * 3 + j] = R[i][j];
#pragma unroll
  for (int k = 0; k < 3; ++k) {
    o[9 + k]  = sumP[k] * invn;  // mu_p (target mean)
    o[12 + k] = sumQ[k] * invn;  // mu_q (mobile mean)
  }
}

// ---------------------------------------------------------------------------
// Pass 3: apply R, per-residue TM term; block reduce -> per-batch TM mean.
// tm = 1/(1 + dev^2/d0^2): no sqrt needed.
// ---------------------------------------------------------------------------
__global__ __launch_bounds__(256) void tm_pass3_score(const float* __restrict__ pred,
                                                      const float* __restrict__ act,
                                                      const float* __restrict__ ws0,
                                                      float* __restrict__ ws2) {
  __shared__ float prm[15];
  __shared__ float red[256];
  const int tid = threadIdx.x;
  const int b   = blockIdx.x;
  if (tid < 15) prm[tid] = ws0[(unsigned)b * 16 + tid];
  __syncthreads();

  const float R00 = prm[0], R01 = prm[1], R02 = prm[2];
  const float R10 = prm[3], R11 = prm[4], R12 = prm[5];
  const float R20 = prm[6], R21 = prm[7], R22 = prm[8];
  const float mp0 = prm[9],  mp1 = prm[10], mp2 = prm[11];
  const float mq0 = prm[12], mq1 = prm[13], mq2 = prm[14];

  const float d0 = 1.24f * cbrtf((float)N_PTS - 15.0f) - 1.8f;
  const float inv_d02 = 1.0f / (d0 * d0);
  const unsigned base = (unsigned)b * (N_PTS * 3);

  float acc = 0.f;
  for (int i = tid; i < N_PTS; i += 256) {
    const unsigned o = base + (unsigned)i * 3;
    float qx = act[o + 0] - mq0;
    float qy = act[o + 1] - mq1;
    float qz = act[o + 2] - mq2;
    // aligned_j = sum_i q_i R[i][j] + mu_p
    float ax = qx * R00 + qy * R10 + qz * R20 + mp0;
    float ay = qx * R01 + qy * R11 + qz * R21 + mp1;
    float az = qx * R02 + qy * R12 + qz * R22 + mp2;
    float dx = pred[o + 0] - ax + EPS_TM;
    float dy = pred[o + 1] - ay + EPS_TM;
    float dz = pred[o + 2] - az + EPS_TM;
    float dev2 = dx * dx + dy * dy + dz * dz;
    acc += 1.0f / (1.0f + dev2 * inv_d02);
  }

  red[tid] = acc;
  __syncthreads();
#pragma unroll
  for (int s = 128; s > 0; s >>= 1) {
    if (tid < s) red[tid] += red[tid + s];
    __syncthreads();
  }
  if (tid == 0) ws2[b] = red[0] * (1.0f / (float)N_PTS);
}

// ---------------------------------------------------------------------------
// Pass 4: reduce per-batch TM means -> scalar loss (negated mean).
// ---------------------------------------------------------------------------
__global__ __launch_bounds__(256) void tm_pass4_reduce(const float* __restrict__ ws2,
                                                       float* __restrict__ out) {
  __shared__ float red[256];
  const int tid = threadIdx.x;
  float s = 0.f;
  for (int i = tid; i < B_BATCH; i += 256) s += ws2[i];
  red[tid] = s;
  __syncthreads();
#pragma unroll
  for (int st = 128; st > 0; st >>= 1) {
    if (tid < st) red[tid] += red[tid + st];
    __syncthreads();
  }
  if (tid == 0) out[0] = -red[0] / (float)B_BATCH;
}

extern "C" void kernel_launch(void* const* d_in, const int* in_sizes, int n_in,
                              void* d_out, int out_size, void* d_ws, size_t ws_size,
                              hipStream_t stream) {
  (void)in_sizes; (void)n_in; (void)out_size; (void)ws_size;
  const float* pred = (const float*)d_in[0];
  const float* act  = (const float*)d_in[1];
  float* out = (float*)d_out;
  float* ws0 = (float*)d_ws;                       // 2048 * 16 floats
  float* ws2 = ws0 + (size_t)B_BATCH * 16;         // 2048 floats

  tm_pass1_cov<<<dim3(B_BATCH), dim3(256), 0, stream>>>(pred, act, ws0);
  tm_pass2_rot<<<dim3(B_BATCH / 256), dim3(256), 0, stream>>>(ws0);
  tm_pass3_score<<<dim3(B_BATCH), dim3(256), 0, stream>>>(pred, act, ws0, ws2);
  tm_pass4_reduce<<<dim3(1), dim3(256), 0, stream>>>(ws2, out);
}